// EdgeLayer_22213570855500
// MI455X (gfx1250) — compile-verified
//
#include <hip/hip_runtime.h>

typedef __attribute__((ext_vector_type(2))) float v2f;
typedef __attribute__((ext_vector_type(8))) float v8f;

#define B_    8
#define N_    4096
#define F_    64
#define K_    20
#define TILE  16
#define PAD   68   // row pitch (floats) for A/B LDS tiles: conflict-free b64 frag reads
#define DPAD  19   // dist tile pitch (floats): conflict-free row-major reads
#define NCAND 40   // merge candidates per row (2 half-lists of 20)

__global__ __launch_bounds__(32)
void edge_knn_kernel(const float* __restrict__ x, float* __restrict__ out) {
  __shared__ float Atile[TILE * PAD];
  __shared__ float Btile[TILE * PAD];
  __shared__ float distT[TILE * DPAD];
  __shared__ float sqRowLds[TILE];
  __shared__ float mergeVals[TILE * NCAND];
  __shared__ int   mergeIdxL[TILE * NCAND];
  __shared__ int   nnIdx[TILE * K_];

  const int lane = threadIdx.x;            // 0..31 (wave32)
  const int wgid = blockIdx.x;             // 0..2047
  const int b    = wgid >> 8;              // N_/TILE = 256 tiles per batch
  const int m0   = (wgid & 255) * TILE;
  const float* xb = x + (size_t)b * N_ * F_;

  const int r16 = lane & 15;
  const int seg = lane >> 4;               // 0 or 1

  // ---- stage A tile (rows m0..m0+15) and compute row norms ----
  float partSq = 0.f;
  {
    const float* src = xb + (size_t)(m0 + r16) * F_ + seg * 32;
    float* dst = Atile + r16 * PAD + seg * 32;
#pragma unroll
    for (int i = 0; i < 8; ++i) {
      float4 v = ((const float4*)src)[i];
      ((float4*)dst)[i] = v;
      partSq += v.x * v.x + v.y * v.y + v.z * v.z + v.w * v.w;
    }
  }
  float rowSq = partSq + __shfl_xor(partSq, 16, 32);
  if (lane < 16) sqRowLds[lane] = rowSq;
  __syncthreads();

  float sumA[8];
#pragma unroll
  for (int r = 0; r < 8; ++r) sumA[r] = sqRowLds[r + 8 * seg];

  // identical frag addressing for A (16x4, M across lanes) and B (4x16, N across lanes)
  const int fragOff = r16 * PAD + seg * 2;

  // ---- hoist loop-invariant A fragments into registers (16 x v2f = 32 VGPRs) ----
  v2f aFrag[F_ / 4];
#pragma unroll
  for (int s = 0; s < F_ / 4; ++s)
    aFrag[s] = *(const v2f*)(Atile + fragOff + 4 * s);

  // ---- per-lane streaming top-K state (row r16, column class seg) ----
  float vals[K_]; int idxs[K_];
#pragma unroll
  for (int t = 0; t < K_; ++t) { vals[t] = __builtin_inff(); idxs[t] = 0x7fffffff; }
  float worstVal = __builtin_inff();
  int   worstIdx = 0x7fffffff;
  int   worstSlot = 0;

  for (int j = 0; j < N_; j += TILE) {
    // ---- stage B tile (cols j..j+15) + column norms ----
    float pB = 0.f;
    {
      const float* src = xb + (size_t)(j + r16) * F_ + seg * 32;
      float* dst = Btile + r16 * PAD + seg * 32;
#pragma unroll
      for (int i = 0; i < 8; ++i) {
        float4 v = ((const float4*)src)[i];
        ((float4*)dst)[i] = v;
        pB += v.x * v.x + v.y * v.y + v.z * v.z + v.w * v.w;
      }
      // prefetch next column tile into cache (global_prefetch_b8)
      if (j + TILE < N_)
        __builtin_prefetch(src + TILE * F_, 0, 0);
    }
    float sqB = pB + __shfl_xor(pB, 16, 32);   // lane's own column norm
    __syncthreads();

    // ---- 16 x V_WMMA_F32_16X16X4_F32 over K=64 (A in regs, B from LDS) ----
    v8f acc = {};
#pragma unroll
    for (int s = 0; s < F_ / 4; ++s) {
      v2f bb = *(const v2f*)(Btile + fragOff + 4 * s);
      acc = __builtin_amdgcn_wmma_f32_16x16x4_f32(false, aFrag[s], false, bb,
                                                  (short)0, acc, false, false);
    }

    // dist = ||a||^2 + ||b||^2 - 2<a,b>; D layout: VGPR r -> M=r+8*seg, N=r16
#pragma unroll
    for (int r = 0; r < 8; ++r) {
      float d = __builtin_fmaf(-2.f, acc[r], sumA[r] + sqB);
      distT[(r + 8 * seg) * DPAD + r16] = d;
    }
    __syncthreads();

    // ---- streaming top-K insert: lane owns row r16, cols seg*8 .. seg*8+7 ----
    const int cbase = seg * 8;
#pragma unroll
    for (int i = 0; i < 8; ++i) {
      float d   = distT[r16 * DPAD + cbase + i];
      int  cidx = j + cbase + i;
      if (d < worstVal) {                 // strict: equal keeps earlier (lower) index
#pragma unroll
        for (int t = 0; t < K_; ++t)
          if (t == worstSlot) { vals[t] = d; idxs[t] = cidx; }
        worstVal = -__builtin_inff(); worstIdx = -1;
#pragma unroll
        for (int t = 0; t < K_; ++t) {
          bool gt = (vals[t] > worstVal) ||
                    (vals[t] == worstVal && idxs[t] > worstIdx);
          if (gt) { worstVal = vals[t]; worstIdx = idxs[t]; worstSlot = t; }
        }
      }
    }
    __syncthreads();
  }

  // ---- merge the two half-lists per row ----
#pragma unroll
  for (int t = 0; t < K_; ++t) {
    mergeVals[r16 * NCAND + seg * K_ + t] = vals[t];
    mergeIdxL[r16 * NCAND + seg * K_ + t] = idxs[t];
  }
  __syncthreads();

  if (lane < 16) {
    const int row = lane;
    for (int k = 0; k < K_; ++k) {        // nearest-first, lower-index tie-break
      float best = __builtin_inff(); int bestIdx = 0x7fffffff; int bestSlot = 0;
      for (int t = 0; t < NCAND; ++t) {
        float v = mergeVals[row * NCAND + t];
        int  ix = mergeIdxL[row * NCAND + t];
        if (v < best || (v == best && ix < bestIdx)) { best = v; bestIdx = ix; bestSlot = t; }
      }
      nnIdx[row * K_ + k] = bestIdx;
      mergeVals[row * NCAND + bestSlot] = __builtin_inff();
    }
  }
  __syncthreads();

  // ---- gather + write edge features: out[b][m][k][0:64]=central, [64:128]=nb-central ----
  for (int row = 0; row < TILE; ++row) {
    const int m = m0 + row;
    float4 c = ((const float4*)(xb + (size_t)m * F_))[r16];
    const size_t obase = (((size_t)b * N_ + m) * K_) * (2 * F_);
    for (int k = 0; k < K_; ++k) {
      int idx = nnIdx[row * K_ + k];
      float4 o;
      if (seg == 0) {
        o = c;
      } else {
        float4 nb = ((const float4*)(xb + (size_t)idx * F_))[r16];
        o = make_float4(nb.x - c.x, nb.y - c.y, nb.z - c.z, nb.w - c.w);
      }
      // lanes 0..31 cover 128 contiguous floats -> fully coalesced 512B segment
      *((float4*)(out + obase + (size_t)k * (2 * F_) + lane * 4)) = o;
    }
  }
}

extern "C" void kernel_launch(void* const* d_in, const int* in_sizes, int n_in,
                              void* d_out, int out_size, void* d_ws, size_t ws_size,
                              hipStream_t stream) {
  (void)in_sizes; (void)n_in; (void)out_size; (void)d_ws; (void)ws_size;
  const float* x = (const float*)d_in[0];
  float* out = (float*)d_out;
  dim3 grid(B_ * (N_ / TILE));   // 2048 single-wave workgroups
  dim3 block(32);
  hipLaunchKernelGGL(edge_knn_kernel, grid, block, 0, stream, x, out);
}